// FusedTensorProductOp4_69295002354279
// MI455X (gfx1250) — compile-verified
//
#include <hip/hip_runtime.h>

// ---- problem constants (match reference) ----
constexpr int kZ  = 20000;
constexpr int kU  = 128;
constexpr int kS  = 8;
constexpr int kP  = 64;
constexpr int kSU = kS * kU;          // 1024
constexpr int kZT = 4;                // z-rows per block
constexpr int kTHREADS = 256;         // 8 waves (wave32)

typedef __attribute__((ext_vector_type(2))) float v2f;
typedef __attribute__((ext_vector_type(4))) float v4f;
typedef __attribute__((ext_vector_type(8))) float v8f;

// out[s,(z,u)] = sum_p W[s,p] * t[p,(z,u)],  W[s,p] = (i3[p]==s) ? coeff[p] : 0
// t[p,(z,u)]  = x0[z,i0[p],u] * x1[z,i1[p],u] * x2[z,i2[p],u]
// Reduction over the 64 paths via 16 chained V_WMMA_F32_16X16X4_F32 (full fp32).
__global__ __launch_bounds__(kTHREADS)
void ftp4_wmma_kernel(const float* __restrict__ x0,
                      const float* __restrict__ x1,
                      const float* __restrict__ x2,
                      const float* __restrict__ pc,
                      const int*   __restrict__ pi,
                      float* __restrict__ out)
{
    __shared__ float lx0[kZT * kSU];
    __shared__ float lx1[kZT * kSU];
    __shared__ float lx2[kZT * kSU];
    __shared__ int4  lpi4[kP];        // one b128 per path: {i0,i1,i2,i3}
    __shared__ float lpc[kP];

    const int tid = threadIdx.x;
    const int z0  = blockIdx.x * kZT;

    // ---- stage 4 z-rows of x0/x1/x2 into LDS (float4, fully coalesced) ----
    {
        const v4f* g0 = (const v4f*)(x0 + (size_t)z0 * kSU);
        const v4f* g1 = (const v4f*)(x1 + (size_t)z0 * kSU);
        const v4f* g2 = (const v4f*)(x2 + (size_t)z0 * kSU);
        v4f* s0 = (v4f*)lx0; v4f* s1 = (v4f*)lx1; v4f* s2 = (v4f*)lx2;
        #pragma unroll
        for (int i = 0; i < (kZT * kSU / 4) / kTHREADS; ++i) {   // 4 iters
            const int idx = tid + i * kTHREADS;
            s0[idx] = g0[idx];
            s1[idx] = g1[idx];
            s2[idx] = g2[idx];
        }
    }
    if (tid < kP) {
        lpi4[tid] = ((const int4*)pi)[tid];
        lpc[tid]  = pc[tid];
    }
    __syncthreads();

    const int wave = tid >> 5;
    const int lane = tid & 31;
    const int h    = lane >> 4;      // half-wave: K-offset pair {0,1} vs {2,3}
    const int n    = lane & 15;      // = A-row M and B-column N for this lane
    const int zl   = wave >> 1;      // 2 waves per z-row
    const int ub   = (wave & 1) * 64;

    const float* bx0 = lx0 + zl * kSU;
    const float* bx1 = lx1 + zl * kSU;
    const float* bx2 = lx2 + zl * kSU;

    v8f c0 = {}; v8f c1 = {}; v8f c2 = {}; v8f c3 = {};

    // K = 64 paths, 4 per WMMA step. Lane owns paths p0 = 4k+2h, p1 = p0+1 in
    // BOTH the A fragment (vgpr0/vgpr1) and the B fragment (vgpr0/vgpr1).
    #pragma unroll
    for (int k = 0; k < 16; ++k) {
        const int p0 = 4 * k + 2 * h;
        const int4 pA = lpi4[p0];
        const int4 pB = lpi4[p0 + 1];
        // Load coefficients unconditionally, then select on the VALUE so the
        // compiler emits v_cndmask (not a predicated ds_load under EXEC masking).
        const float cA = lpc[p0];
        const float cB = lpc[p0 + 1];
        v2f a;
        a.x = (pA.w == n) ? cA : 0.0f;   // i3 in [0,8) => pad rows 8..15 auto-zero
        a.y = (pB.w == n) ? cB : 0.0f;

        const float* a0 = bx0 + pA.x * kU;
        const float* a1 = bx1 + pA.y * kU;
        const float* a2 = bx2 + pA.z * kU;
        const float* b0 = bx0 + pB.x * kU;
        const float* b1 = bx1 + pB.y * kU;
        const float* b2 = bx2 + pB.z * kU;

        #pragma unroll
        for (int t = 0; t < 4; ++t) {
            const int u = ub + t * 16 + n;
            v2f b;
            b.x = a0[u] * a1[u] * a2[u];
            b.y = b0[u] * b1[u] * b2[u];
            v8f& c = (t == 0) ? c0 : (t == 1) ? c1 : (t == 2) ? c2 : c3;
            // 8 args: (neg_a, A, neg_b, B, c_mod, C, reuse_a, reuse_b)
            c = __builtin_amdgcn_wmma_f32_16x16x4_f32(
                    false, a, false, b, (short)0, c, false, false);
        }
    }

    // D layout: lanes 0-15 hold M=vgpr (rows 0..7 = the 8 output slots), N=lane.
    // Lanes 16-31 hold pad rows 8..15 (all zero) -> nothing to store.
    if (h == 0) {
        float* o = out + (size_t)(z0 + zl) * kSU;
        #pragma unroll
        for (int m = 0; m < kS; ++m) {
            o[m * kU + ub +  0 + n] = c0[m];
            o[m * kU + ub + 16 + n] = c1[m];
            o[m * kU + ub + 32 + n] = c2[m];
            o[m * kU + ub + 48 + n] = c3[m];
        }
    }
}

extern "C" void kernel_launch(void* const* d_in, const int* in_sizes, int n_in,
                              void* d_out, int out_size, void* d_ws, size_t ws_size,
                              hipStream_t stream) {
    const float* x0 = (const float*)d_in[0];
    const float* x1 = (const float*)d_in[1];
    const float* x2 = (const float*)d_in[2];
    const float* pc = (const float*)d_in[3];
    const int*   pi = (const int*)d_in[4];
    float* out = (float*)d_out;

    dim3 grid(kZ / kZT);      // 5000 blocks, Z divisible by 4
    dim3 block(kTHREADS);
    hipLaunchKernelGGL(ftp4_wmma_kernel, grid, block, 0, stream,
                       x0, x1, x2, pc, pi, out);
}